// GAT_39436389712499
// MI455X (gfx1250) — compile-verified
//
#include <hip/hip_runtime.h>
#include <hip/hip_bf16.h>

#define NNODES 100000
#define NEDGES 800000
#define HC 128
#define NH 4

typedef float v2f __attribute__((ext_vector_type(2)));
typedef float v8f __attribute__((ext_vector_type(8)));

static inline int cdiv(int a, int b) { return (a + b - 1) / b; }

// ---------------- init: zero accumulator, set segment-max to -inf, segment-sum to 0
__global__ __launch_bounds__(256) void k_init_layer(float* __restrict__ acc, int nacc,
                                                    float* __restrict__ m,
                                                    float* __restrict__ s, int nm) {
  int i = blockIdx.x * blockDim.x + threadIdx.x;
  if (i < nacc) acc[i] = 0.0f;
  if (i < nm) {
    m[i] = __int_as_float(0xff800000);  // -inf
    s[i] = 0.0f;
  }
}

// ---------------- fp32 WMMA GEMM: XP[rows,128] = X[rows,K] * W[128,K]^T
// One wave computes a 16x128 strip: 8 accumulators of 16x16, K-step 4,
// A fragment (x rows) reused across all 8 column tiles.
__global__ __launch_bounds__(256) void k_gemm_wmma(const float* __restrict__ X,
                                                   const float* __restrict__ W,
                                                   float* __restrict__ XP,
                                                   int rows, int K) {
  int wave = (blockIdx.x * blockDim.x + threadIdx.x) >> 5;
  int lane = threadIdx.x & 31;
  int m0 = wave * 16;
  if (m0 >= rows) return;  // wave-uniform: EXEC stays all-ones for WMMA

  int n15 = lane & 15;            // A: row within tile; B: col within tile
  int khalf = (lane >> 4) * 2;    // lanes 16-31 hold K+2,K+3

  v8f acc[8] = {};
  const float* xbase = X + (size_t)(m0 + n15) * K + khalf;
  const float* wbase = W + (size_t)n15 * K + khalf;

  for (int k = 0; k < K; k += 4) {
    v2f a = *(const v2f*)(xbase + k);
#pragma unroll
    for (int nt = 0; nt < 8; ++nt) {
      v2f b = *(const v2f*)(wbase + (size_t)nt * 16 * K + k);
      acc[nt] = __builtin_amdgcn_wmma_f32_16x16x4_f32(
          /*neg_a=*/false, a, /*neg_b=*/false, b,
          /*c_mod=*/(short)0, acc[nt], /*reuse_a=*/false, /*reuse_b=*/false);
    }
  }

  int half = lane >> 4;
#pragma unroll
  for (int nt = 0; nt < 8; ++nt) {
#pragma unroll
    for (int i = 0; i < 8; ++i) {
      int m = m0 + half * 8 + i;  // C/D layout: VGPR i -> M = i (+8 for hi lanes)
      XP[(size_t)m * HC + nt * 16 + n15] = acc[nt][i];
    }
  }
}

// ---------------- per-(node,head) attention dot products
__global__ __launch_bounds__(256) void k_attn_dots(const float* __restrict__ XP,
                                                   const float* __restrict__ att_src,
                                                   const float* __restrict__ att_dst,
                                                   float* __restrict__ ASRC,
                                                   float* __restrict__ ADST, int n) {
  int idx = blockIdx.x * blockDim.x + threadIdx.x;
  if (idx >= n * NH) return;
  int node = idx >> 2;
  int h = idx & 3;
  const float* xp = XP + (size_t)node * HC + h * 32;
  const float* as = att_src + h * 32;
  const float* ad = att_dst + h * 32;
  float s0 = 0.f, s1 = 0.f;
#pragma unroll
  for (int c = 0; c < 32; ++c) {
    float v = xp[c];
    s0 += v * as[c];
    s1 += v * ad[c];
  }
  ASRC[idx] = s0;
  ADST[idx] = s1;
}

// float atomic-max via ordered-int trick (init value must be -inf)
__device__ __forceinline__ void atomicMaxF(float* addr, float val) {
  if (val >= 0.0f)
    atomicMax((int*)addr, __float_as_int(val));
  else
    atomicMin((unsigned int*)addr, __float_as_uint(val));
}

// ---------------- edge logits: leaky_relu(a_src[src]+a_dst[dst]) + segment max
__global__ __launch_bounds__(256) void k_edge_logits(const int* __restrict__ src,
                                                     const int* __restrict__ dst,
                                                     const float* __restrict__ ASRC,
                                                     const float* __restrict__ ADST,
                                                     float* __restrict__ EB,
                                                     float* __restrict__ M, int ne) {
  int e = blockIdx.x * blockDim.x + threadIdx.x;
  if (e >= ne) return;
  int s = src[e], d = dst[e];
#pragma unroll
  for (int h = 0; h < NH; ++h) {
    float v = ASRC[(size_t)s * NH + h] + ADST[(size_t)d * NH + h];
    v = v > 0.f ? v : 0.2f * v;  // leaky relu, slope 0.2
    EB[(size_t)e * NH + h] = v;
    atomicMaxF(&M[(size_t)d * NH + h], v);
  }
}

// ---------------- exp(e - max) + segment sum (EB rewritten in place with ex)
__global__ __launch_bounds__(256) void k_edge_exp(const int* __restrict__ dst,
                                                  float* __restrict__ EB,
                                                  const float* __restrict__ M,
                                                  float* __restrict__ S, int ne) {
  int e = blockIdx.x * blockDim.x + threadIdx.x;
  if (e >= ne) return;
  int d = dst[e];
#pragma unroll
  for (int h = 0; h < NH; ++h) {
    float ex = __expf(EB[(size_t)e * NH + h] - M[(size_t)d * NH + h]);
    EB[(size_t)e * NH + h] = ex;
    atomicAdd(&S[(size_t)d * NH + h], ex);
  }
}

// ---------------- aggregate: out[dst] += xp[src] * alpha. One wave per edge,
// each lane handles 4 consecutive features (float4 gather + 4 f32 atomics).
__global__ __launch_bounds__(256) void k_edge_aggr(const int* __restrict__ src,
                                                   const int* __restrict__ dst,
                                                   const float* __restrict__ XP,
                                                   const float* __restrict__ EB,
                                                   const float* __restrict__ S,
                                                   float* __restrict__ OUT, int ne) {
  int wave = (blockIdx.x * blockDim.x + threadIdx.x) >> 5;
  int lane = threadIdx.x & 31;
  if (wave >= ne) return;
  int s = src[wave], d = dst[wave];
  int c0 = lane * 4;
  int h = lane >> 3;  // 4 feats per lane all inside one 32-wide head chunk
  float alpha = EB[(size_t)wave * NH + h] / (S[(size_t)d * NH + h] + 1e-16f);
  const float4 v = *(const float4*)(XP + (size_t)s * HC + c0);
  float* o = OUT + (size_t)d * HC + c0;
  atomicAdd(o + 0, v.x * alpha);
  atomicAdd(o + 1, v.y * alpha);
  atomicAdd(o + 2, v.z * alpha);
  atomicAdd(o + 3, v.w * alpha);
}

// ---------------- bias (+ optional relu), in place
__global__ __launch_bounds__(256) void k_bias_act(float* __restrict__ buf,
                                                  const float* __restrict__ bias,
                                                  int n, int relu) {
  int i = blockIdx.x * blockDim.x + threadIdx.x;
  if (i >= n) return;
  float v = buf[i] + bias[i & (HC - 1)];
  if (relu) v = fmaxf(v, 0.0f);
  buf[i] = v;
}

extern "C" void kernel_launch(void* const* d_in, const int* in_sizes, int n_in,
                              void* d_out, int out_size, void* d_ws, size_t ws_size,
                              hipStream_t stream) {
  const float* x     = (const float*)d_in[0];
  const int*   ei    = (const int*)d_in[1];
  const float* W0    = (const float*)d_in[2];
  const float* atts0 = (const float*)d_in[3];
  const float* attd0 = (const float*)d_in[4];
  const float* b0    = (const float*)d_in[5];
  const float* W1    = (const float*)d_in[6];
  const float* atts1 = (const float*)d_in[7];
  const float* attd1 = (const float*)d_in[8];
  const float* b1    = (const float*)d_in[9];

  const int n = NNODES, ne = NEDGES;
  const int* srcv = ei;
  const int* dstv = ei + ne;

  char* ws = (char*)d_ws;
  float* XP   = (float*)(ws + 0);          // n*128 f32 = 51.2 MB
  float* H0   = (float*)(ws + 51200000);   // n*128 f32 = 51.2 MB
  float* ASRC = (float*)(ws + 102400000);  // n*4
  float* ADST = (float*)(ws + 104000000);  // n*4
  float* M    = (float*)(ws + 105600000);  // n*4
  float* S    = (float*)(ws + 107200000);  // n*4
  float* EB   = (float*)(ws + 108800000);  // ne*4 = 12.8 MB
  float* OUT  = (float*)d_out;

  const dim3 blk(256);
  const int nAcc = n * HC;   // 12.8M
  const int nMH  = n * NH;   // 400k
  const int gemmBlocks = cdiv(cdiv(n, 16) * 32, 256);

  // ---------- layer 0 ----------
  k_init_layer<<<cdiv(nAcc, 256), blk, 0, stream>>>(H0, nAcc, M, S, nMH);
  k_gemm_wmma<<<gemmBlocks, blk, 0, stream>>>(x, W0, XP, n, 256);
  k_attn_dots<<<cdiv(nMH, 256), blk, 0, stream>>>(XP, atts0, attd0, ASRC, ADST, n);
  k_edge_logits<<<cdiv(ne, 256), blk, 0, stream>>>(srcv, dstv, ASRC, ADST, EB, M, ne);
  k_edge_exp<<<cdiv(ne, 256), blk, 0, stream>>>(dstv, EB, M, S, ne);
  k_edge_aggr<<<cdiv(ne * 32, 256), blk, 0, stream>>>(srcv, dstv, XP, EB, S, H0, ne);
  k_bias_act<<<cdiv(nAcc, 256), blk, 0, stream>>>(H0, b0, nAcc, /*relu=*/1);

  // ---------- layer 1 ----------
  k_init_layer<<<cdiv(nAcc, 256), blk, 0, stream>>>(OUT, nAcc, M, S, nMH);
  k_gemm_wmma<<<gemmBlocks, blk, 0, stream>>>(H0, W1, XP, n, 128);
  k_attn_dots<<<cdiv(nMH, 256), blk, 0, stream>>>(XP, atts1, attd1, ASRC, ADST, n);
  k_edge_logits<<<cdiv(ne, 256), blk, 0, stream>>>(srcv, dstv, ASRC, ADST, EB, M, ne);
  k_edge_exp<<<cdiv(ne, 256), blk, 0, stream>>>(dstv, EB, M, S, ne);
  k_edge_aggr<<<cdiv(ne * 32, 256), blk, 0, stream>>>(srcv, dstv, XP, EB, S, OUT, ne);
  k_bias_act<<<cdiv(nAcc, 256), blk, 0, stream>>>(OUT, b1, nAcc, /*relu=*/0);
}